// TransductiveLayer_601295422043
// MI455X (gfx1250) — compile-verified
//
#include <hip/hip_runtime.h>

typedef __attribute__((ext_vector_type(2))) float v2f;
typedef __attribute__((ext_vector_type(8))) float v8f;

#define N_NODES 100000
#define N_EDGES 1600000
#define N_HOPS  3
#define IN_DIM  256
#define OUT_DIM 128
#define LDS_PITCH (IN_DIM + 4)  // 260 dwords: +4 banks/row -> conflict-free A-frag reads
#define ROW_TILES 3             // 48 rows per block; LDS = 3*16*260*4 = 49.9 KB (< 64 KB static limit)
#define ROWS_PER_BLOCK (ROW_TILES * 16)

__device__ __forceinline__ void atomic_add_f32(float* p, float v) {
    // relaxed, agent scope -> no-return global_atomic_add_f32 path (STOREcnt-tracked)
    __hip_atomic_fetch_add(p, v, __ATOMIC_RELAXED, __HIP_MEMORY_SCOPE_AGENT);
}

// ---------------------------------------------------------------------------
// h = X @ W   (X: [N,256] f32, W: [256,128] f32, h: [N,128] f32)
// Block = 48 output rows (3 x 16-row tiles); 8 waves x 16 cols = 128 cols.
// Each wave keeps 3 v8f accumulators so every B fragment feeds 3 WMMAs.
// v_wmma_f32_16x16x4_f32, K stepped by 4 (64 K-steps, 192 WMMAs per wave).
// N_NODES % 16 == 0, so each 16-row tile is wholly valid or wholly invalid:
// the tail guard is a uniform scalar branch per tile (no EXEC churn).
// ---------------------------------------------------------------------------
__global__ __launch_bounds__(256) void gemm_wmma_kernel(const float* __restrict__ X,
                                                        const float* __restrict__ W,
                                                        float* __restrict__ H) {
    __shared__ float xs[ROWS_PER_BLOCK * LDS_PITCH];

    const int r0  = blockIdx.x * ROWS_PER_BLOCK;
    const int tid = threadIdx.x;

    // Cooperative load of the 48x256 X stripe (48 elements/thread, coalesced).
    // Clamp the row index for the overhang block: no EXEC divergence here.
#pragma unroll
    for (int i = 0; i < ROWS_PER_BLOCK; ++i) {
        const int idx = tid + i * 256;
        const int row = idx >> 8;       // /256
        const int col = idx & 255;
        int grow = r0 + row;
        grow = grow < N_NODES ? grow : (N_NODES - 1);
        xs[row * LDS_PITCH + col] = X[(size_t)grow * IN_DIM + col];
    }
    __syncthreads();

    const int wave = tid >> 5;
    const int lane = tid & 31;
    const int c0   = wave * 16;
    const int m    = lane & 15;        // row within tile (A) / col within tile (B,C)
    const int kh   = (lane >> 4) * 2;  // lanes 16-31 carry K+2,K+3

    const float* wcol = W + c0 + m;    // this lane's column of W for the B-frag

    v8f acc0 = {}, acc1 = {}, acc2 = {};
#pragma unroll 4
    for (int ks = 0; ks < IN_DIM; ks += 4) {
        const int kb = ks + kh;
        v2f b;
        // B 4x16: lanes stripe N, VGPRs stripe K with the lane-half K split
        b.x = wcol[kb * OUT_DIM];
        b.y = wcol[(kb + 1) * OUT_DIM];

        // 32-bit A 16x4 layout: lanes 0-15 -> K=kb,kb+1 ; lanes 16-31 -> K=kb,kb+1 = ks+2,ks+3
        v2f a0, a1, a2;
        const float* xr0 = xs + (0 * 16 + m) * LDS_PITCH + kb;
        const float* xr1 = xs + (1 * 16 + m) * LDS_PITCH + kb;
        const float* xr2 = xs + (2 * 16 + m) * LDS_PITCH + kb;
        a0.x = xr0[0]; a0.y = xr0[1];
        a1.x = xr1[0]; a1.y = xr1[1];
        a2.x = xr2[0]; a2.y = xr2[1];

        acc0 = __builtin_amdgcn_wmma_f32_16x16x4_f32(false, a0, false, b, (short)0, acc0, false, false);
        acc1 = __builtin_amdgcn_wmma_f32_16x16x4_f32(false, a1, false, b, (short)0, acc1, false, false);
        acc2 = __builtin_amdgcn_wmma_f32_16x16x4_f32(false, a2, false, b, (short)0, acc2, false, false);
    }

    // 16x16 f32 C/D layout: VGPR i, lanes 0-15 -> M=i ; lanes 16-31 -> M=i+8.
    // Uniform (scalar) validity branch per tile; straight-line b32 stores inside.
    const int row_hi = (lane >> 4) * 8;

    if (r0 + 0 * 16 < N_NODES) {
        float* hp = H + (size_t)(r0 + 0 * 16 + row_hi) * OUT_DIM + c0 + m;
#pragma unroll
        for (int i = 0; i < 8; ++i) hp[i * OUT_DIM] = acc0[i];
    }
    if (r0 + 1 * 16 < N_NODES) {
        float* hp = H + (size_t)(r0 + 1 * 16 + row_hi) * OUT_DIM + c0 + m;
#pragma unroll
        for (int i = 0; i < 8; ++i) hp[i * OUT_DIM] = acc1[i];
    }
    if (r0 + 2 * 16 < N_NODES) {
        float* hp = H + (size_t)(r0 + 2 * 16 + row_hi) * OUT_DIM + c0 + m;
#pragma unroll
        for (int i = 0; i < 8; ++i) hp[i * OUT_DIM] = acc2[i];
    }
}

// ---------------------------------------------------------------------------
// out[dst] += val * h[src]  -- one wave32 per edge, float4 per lane (128 cols)
// ---------------------------------------------------------------------------
__global__ __launch_bounds__(256) void edge_scatter_kernel(const float* __restrict__ H,
                                                           const int* __restrict__ src,
                                                           const int* __restrict__ dst,
                                                           const float* __restrict__ vals,
                                                           float* __restrict__ out) {
    const int e    = blockIdx.x * 8 + (threadIdx.x >> 5);
    const int lane = threadIdx.x & 31;
    if (e >= N_EDGES) return;

    const int   s = src[e];
    const int   d = dst[e];
    const float v = vals[e];

    const float4 h4 = ((const float4*)(H + (size_t)s * OUT_DIM))[lane];
    float* op = out + (size_t)d * OUT_DIM + lane * 4;
    atomic_add_f32(op + 0, v * h4.x);
    atomic_add_f32(op + 1, v * h4.y);
    atomic_add_f32(op + 2, v * h4.z);
    atomic_add_f32(op + 3, v * h4.w);
}

// ---------------------------------------------------------------------------
// Init / finalize
// ---------------------------------------------------------------------------
__global__ __launch_bounds__(256) void zero_kernel(float4* __restrict__ out, int n4) {
    const int i = blockIdx.x * blockDim.x + threadIdx.x;
    if (i < n4) out[i] = make_float4(0.f, 0.f, 0.f, 0.f);
}

__global__ __launch_bounds__(256) void relu_kernel(float4* __restrict__ out, int n4) {
    const int i = blockIdx.x * blockDim.x + threadIdx.x;
    if (i < n4) {
        float4 v = out[i];
        v.x = fmaxf(v.x, 0.f);
        v.y = fmaxf(v.y, 0.f);
        v.z = fmaxf(v.z, 0.f);
        v.w = fmaxf(v.w, 0.f);
        out[i] = v;
    }
}

extern "C" void kernel_launch(void* const* d_in, const int* in_sizes, int n_in,
                              void* d_out, int out_size, void* d_ws, size_t ws_size,
                              hipStream_t stream) {
    const float* X     = (const float*)d_in[0];  // [100000, 256]
    const float* W     = (const float*)d_in[1];  // [3, 256, 128]
    const int*   esrc  = (const int*)  d_in[2];  // [3, 1600000]
    const int*   edst  = (const int*)  d_in[3];  // [3, 1600000]
    const float* evals = (const float*)d_in[4];  // [3, 1600000]
    float*       out   = (float*)d_out;          // [100000, 128]
    float*       H     = (float*)d_ws;           // 100000*128*4 = 51.2 MB scratch

    const int n4 = N_NODES * OUT_DIM / 4;        // 3,200,000 float4
    zero_kernel<<<(n4 + 255) / 256, 256, 0, stream>>>((float4*)out, n4);

    const int gemm_blocks = (N_NODES + ROWS_PER_BLOCK - 1) / ROWS_PER_BLOCK;  // 2084
    for (int hop = 0; hop < N_HOPS; ++hop) {
        gemm_wmma_kernel<<<gemm_blocks, 256, 0, stream>>>(
            X, W + (size_t)hop * IN_DIM * OUT_DIM, H);
        edge_scatter_kernel<<<N_EDGES / 8, 256, 0, stream>>>(
            H, esrc + (size_t)hop * N_EDGES, edst + (size_t)hop * N_EDGES,
            evals + (size_t)hop * N_EDGES, out);
    }

    relu_kernel<<<(n4 + 255) / 256, 256, 0, stream>>>((float4*)out, n4);
}